// GCN_22316650070136
// MI455X (gfx1250) — compile-verified
//
#include <hip/hip_runtime.h>

typedef __attribute__((ext_vector_type(2))) float v2f;
typedef __attribute__((ext_vector_type(8))) float v8f;

#define N_NODES  100000
#define N_EDGES  1600000
#define N_GRAPHS 64
#define IN_DIM   128
#define HIDDEN   64
#define OUT_DIM  6

// ---------------- utility kernels ----------------

__global__ __launch_bounds__(256) void zero_kernel(float* __restrict__ p, int n) {
    int i = blockIdx.x * blockDim.x + threadIdx.x;
    if (i < n) p[i] = 0.0f;
}

__global__ __launch_bounds__(256) void deg_kernel(const int* __restrict__ dst,
                                                  float* __restrict__ deg) {
    int e = blockIdx.x * blockDim.x + threadIdx.x;
    if (e < N_EDGES) atomicAdd(&deg[dst[e]], 1.0f);
}

__global__ __launch_bounds__(256) void dinv_kernel(float* __restrict__ deg) {
    int i = blockIdx.x * blockDim.x + threadIdx.x;
    if (i < N_NODES) deg[i] = rsqrtf(deg[i] + 1.0f);   // deg incl. self-loop, in-place -> dinv
}

// ---------------- WMMA fp32 GEMM: Y[nrows,64] = act(X[nrows,K]) @ W[K,64] ----------------
// One wave computes a 16x64 tile with V_WMMA_F32_16X16X4_F32.
// A (16x4 f32): lanes 0-15 hold row M=lane, VGPR0=K0,VGPR1=K1; lanes 16-31 K2,K3 -> float2 load.
// B (4x16 f32): VGPR0 = row {k0 | k0+2}, VGPR1 = row {k0+1 | k0+3}, col = lane&15.
// C/D (16x16): VGPR j -> row j (lanes 0-15) / row j+8 (lanes 16-31), col = lane&15.
template <int K, bool RELU>
__global__ __launch_bounds__(128) void gemm_wmma_f32(const float* __restrict__ X,
                                                     const float* __restrict__ W,
                                                     float* __restrict__ Y,
                                                     int nrows) {
    const int lane = threadIdx.x & 31;
    const int wave = threadIdx.x >> 5;
    const int rowbase = (blockIdx.x * 4 + wave) * 16;
    if (rowbase >= nrows) return;

    const int lr = lane & 15;   // row-in-tile for A, col-in-tile for B/C
    const int hl = lane >> 4;   // half-wave select

    v8f acc0 = {}, acc1 = {}, acc2 = {}, acc3 = {};

    const float* xrow = X + (rowbase + lr) * K + hl * 2;

    for (int k0 = 0; k0 < K; k0 += 4) {
        v2f a = *(const v2f*)(xrow + k0);
        if (RELU) { a.x = fmaxf(a.x, 0.0f); a.y = fmaxf(a.y, 0.0f); }

        const float* wp = W + (k0 + hl * 2) * HIDDEN + lr;
        v2f b0; b0.x = wp[0];  b0.y = wp[HIDDEN + 0];
        v2f b1; b1.x = wp[16]; b1.y = wp[HIDDEN + 16];
        v2f b2; b2.x = wp[32]; b2.y = wp[HIDDEN + 32];
        v2f b3; b3.x = wp[48]; b3.y = wp[HIDDEN + 48];

        acc0 = __builtin_amdgcn_wmma_f32_16x16x4_f32(false, a, false, b0, (short)0, acc0, false, false);
        acc1 = __builtin_amdgcn_wmma_f32_16x16x4_f32(false, a, false, b1, (short)0, acc1, false, false);
        acc2 = __builtin_amdgcn_wmma_f32_16x16x4_f32(false, a, false, b2, (short)0, acc2, false, false);
        acc3 = __builtin_amdgcn_wmma_f32_16x16x4_f32(false, a, false, b3, (short)0, acc3, false, false);
    }

    float* yrow = Y + (rowbase + hl * 8) * HIDDEN + lr;
#pragma unroll
    for (int j = 0; j < 8; ++j) {
        yrow[j * HIDDEN + 0]  = acc0[j];
        yrow[j * HIDDEN + 16] = acc1[j];
        yrow[j * HIDDEN + 32] = acc2[j];
        yrow[j * HIDDEN + 48] = acc3[j];
    }
}

// ---------------- scatter-buffer init: B[i] = xw[i]*dinv[node]^2 + bias[f] ----------------
// (self-loop contribution + bias folded into the accumulator init, replacing a zero pass)

__global__ __launch_bounds__(256) void init_agg_kernel(const float* __restrict__ xw,
                                                       const float* __restrict__ dinv,
                                                       const float* __restrict__ bias,
                                                       float* __restrict__ agg) {
    int i = blockIdx.x * blockDim.x + threadIdx.x;
    if (i >= N_NODES * HIDDEN) return;
    int node = i >> 6;
    int f = i & 63;
    float dv = dinv[node];
    agg[i] = xw[i] * dv * dv + bias[f];
}

// ---------------- edge gather/scatter: agg[dst] += xw[src] * dinv[src]*dinv[dst] ----------------

__global__ __launch_bounds__(256) void edge_agg_kernel(const int* __restrict__ src,
                                                       const int* __restrict__ dst,
                                                       const float* __restrict__ dinv,
                                                       const float* __restrict__ xw,
                                                       float* __restrict__ agg) {
    int gid = blockIdx.x * blockDim.x + threadIdx.x;   // (edge, feature) pair
    int e = gid >> 6;
    int f = gid & 63;
    if (e >= N_EDGES) return;
    int s = src[e];
    int d = dst[e];
    float coef = dinv[s] * dinv[d];
    atomicAdd(&agg[d * HIDDEN + f], xw[s * HIDDEN + f] * coef);
}

// ---------------- graph mean-pool accumulation (layer-2 output is already complete in h2) ----------------

__global__ __launch_bounds__(256) void pool_kernel(const float* __restrict__ h2,
                                                   const int* __restrict__ batch,
                                                   float* __restrict__ sums,
                                                   float* __restrict__ cnt) {
    int i = blockIdx.x * blockDim.x + threadIdx.x;
    if (i >= N_NODES * HIDDEN) return;
    int node = i >> 6;
    int f = i & 63;
    int g = batch[node];
    atomicAdd(&sums[g * HIDDEN + f], h2[i]);
    if (f == 0) atomicAdd(&cnt[g], 1.0f);
}

// ---------------- head: out[g,o] = (sums[g]/cnt[g]) @ Wl + bl ----------------

__global__ __launch_bounds__(384) void head_kernel(const float* __restrict__ sums,
                                                   const float* __restrict__ cnt,
                                                   const float* __restrict__ Wl,
                                                   const float* __restrict__ bl,
                                                   float* __restrict__ out) {
    int t = threadIdx.x;
    if (t >= N_GRAPHS * OUT_DIM) return;
    int g = t / OUT_DIM;
    int o = t % OUT_DIM;
    float inv_c = 1.0f / fmaxf(cnt[g], 1.0f);
    float acc = bl[o];
#pragma unroll 8
    for (int f = 0; f < HIDDEN; ++f)
        acc += (sums[g * HIDDEN + f] * inv_c) * Wl[f * OUT_DIM + o];
    out[t] = acc;
}

// ---------------- launch ----------------

extern "C" void kernel_launch(void* const* d_in, const int* in_sizes, int n_in,
                              void* d_out, int out_size, void* d_ws, size_t ws_size,
                              hipStream_t stream) {
    const float* x     = (const float*)d_in[0];
    const int*   edge  = (const int*)d_in[1];      // edge_index flat [2, E]
    const int*   batch = (const int*)d_in[2];
    const float* W1    = (const float*)d_in[3];
    const float* b1    = (const float*)d_in[4];
    const float* W2    = (const float*)d_in[5];
    const float* b2    = (const float*)d_in[6];
    const float* Wl    = (const float*)d_in[7];
    const float* bl    = (const float*)d_in[8];
    float* out = (float*)d_out;

    const int* src = edge;
    const int* dst = edge + N_EDGES;

    const int NF = N_NODES * HIDDEN;               // 6.4M floats

    float* A    = (float*)d_ws;                    // xw / hw  (25.6 MB)
    float* B    = A + NF;                          // pre-act h / h2 accumulator (25.6 MB)
    float* dinv = B + NF;                          // 100k floats
    float* sums = dinv + N_NODES;                  // 64*64
    float* cnt  = sums + N_GRAPHS * HIDDEN;        // 64

    const int gemm_blocks = (N_NODES / 16 + 3) / 4;      // 4 waves/block, 16 rows/wave
    const int nf_blocks   = (NF + 255) / 256;
    const int edge_blocks = (N_EDGES * 64) / 256;        // exact: 400000

    // degrees -> dinv
    zero_kernel<<<(N_NODES + 255) / 256, 256, 0, stream>>>(dinv, N_NODES);
    deg_kernel<<<(N_EDGES + 255) / 256, 256, 0, stream>>>(dst, dinv);
    dinv_kernel<<<(N_NODES + 255) / 256, 256, 0, stream>>>(dinv);

    // layer 1: A = x @ W1 ; B = A*dinv^2 + b1 ; B += scatter(A)   (pre-activation h)
    gemm_wmma_f32<IN_DIM, false><<<gemm_blocks, 128, 0, stream>>>(x, W1, A, N_NODES);
    init_agg_kernel<<<nf_blocks, 256, 0, stream>>>(A, dinv, b1, B);
    edge_agg_kernel<<<edge_blocks, 256, 0, stream>>>(src, dst, dinv, A, B);

    // layer 2: A = relu(B) @ W2 (ReLU fused into GEMM A-load) ; B = A*dinv^2 + b2 ; B += scatter(A)
    gemm_wmma_f32<HIDDEN, true><<<gemm_blocks, 128, 0, stream>>>(B, W2, A, N_NODES);
    init_agg_kernel<<<nf_blocks, 256, 0, stream>>>(A, dinv, b2, B);
    edge_agg_kernel<<<edge_blocks, 256, 0, stream>>>(src, dst, dinv, A, B);

    // mean-pool + head
    zero_kernel<<<(N_GRAPHS * HIDDEN + N_GRAPHS + 255) / 256, 256, 0, stream>>>(sums, N_GRAPHS * HIDDEN + N_GRAPHS);
    pool_kernel<<<nf_blocks, 256, 0, stream>>>(B, batch, sums, cnt);
    head_kernel<<<1, 384, 0, stream>>>(sums, cnt, Wl, bl, out);
}